// GIN_14078902796336
// MI455X (gfx1250) — compile-verified
//
#include <hip/hip_runtime.h>

typedef float v2f __attribute__((ext_vector_type(2)));
typedef float v8f __attribute__((ext_vector_type(8)));

#define HDIM 128

// ---------------------------------------------------------------------------
// Fused GEMM: Out[m,n] = act( ((A[m,:] (+A2[m,:])) dot W[n,:]) + bias[n] )
// optional per-col affine (BN eval mode): val = val*scale[n]*scale_const + shift[n]
// One block = 256 threads = 8 wave32s; each wave computes a 16-row x 128-col
// stripe via V_WMMA_F32_16X16X4_F32. W (128x128 f32, 64KB) staged in LDS.
// Template-specialized so the K-loop and epilogue are branch-free.
// ---------------------------------------------------------------------------
template <bool HAS_A2, bool HAS_BN, bool RELU>
__global__ __launch_bounds__(256) void gemm128_wmma(
    const float* __restrict__ A,     // [nrows,128]
    const float* __restrict__ A2,    // elementwise add (used iff HAS_A2)
    const float* __restrict__ W,     // [128,128] row-major; Out = A @ W^T
    const float* __restrict__ bias,  // [128]
    const float* __restrict__ scale, // [128] (used iff HAS_BN)
    const float* __restrict__ shift, // [128] (used iff HAS_BN)
    float* __restrict__ Out,         // [nrows,128]
    int nrows, float scale_const)
{
    __shared__ float Wlds[HDIM * HDIM]; // 64 KB of the 320 KB/WGP LDS

    const int tid = threadIdx.x;
    // Cooperative 64KB weight stage (float4, coalesced).
    {
        const float4* Wv = (const float4*)W;
        float4*       Lv = (float4*)Wlds;
        #pragma unroll
        for (int i = 0; i < (HDIM * HDIM / 4) / 256; ++i)
            Lv[tid + i * 256] = Wv[tid + i * 256];
    }
    __syncthreads();

    const int wave  = tid >> 5;        // 0..7
    const int lane  = tid & 31;
    const int lrow  = lane & 15;       // M (A) / N (B,C,D) within tile
    const int khalf = lane >> 4;       // which K-pair half

    const int row_base = blockIdx.x * 128 + wave * 16;
    const int row      = row_base + lrow;
    const int arow     = row < nrows ? row : (nrows - 1); // clamp: EXEC stays all-1s

    const float* Arow  = A + (size_t)arow * HDIM;
    const float* A2row = HAS_A2 ? (A2 + (size_t)arow * HDIM) : nullptr;

    v8f acc[8];
    const v8f vzero = {0.f, 0.f, 0.f, 0.f, 0.f, 0.f, 0.f, 0.f};
    #pragma unroll
    for (int t = 0; t < 8; ++t) acc[t] = vzero;

    // K loop: 32 steps of K=4; 8 column tiles per wave -> 256 WMMAs/wave.
    #pragma unroll 4
    for (int kb = 0; kb < HDIM; kb += 4) {
        const int k0 = kb + khalf * 2;
        v2f afrag;
        afrag.x = Arow[k0];
        afrag.y = Arow[k0 + 1];
        if constexpr (HAS_A2) {
            afrag.x += A2row[k0];
            afrag.y += A2row[k0 + 1];
        }
        #pragma unroll
        for (int ct = 0; ct < 8; ++ct) {
            const int n = ct * 16 + lrow;      // B = W^T: B[k][n] = W[n][k]
            v2f bfrag;
            bfrag.x = Wlds[n * HDIM + k0];
            bfrag.y = Wlds[n * HDIM + k0 + 1];
            acc[ct] = __builtin_amdgcn_wmma_f32_16x16x4_f32(
                false, afrag, false, bfrag, (short)0, acc[ct], false, false);
        }
    }

    // Epilogue: lane l, VGPR v -> M = 8*(l>>4)+v, N = ct*16 + (l&15).
    // Wave-uniform full-tile fast path: only the final tile takes guarded stores.
    if (row_base + 16 <= nrows) {
        #pragma unroll
        for (int ct = 0; ct < 8; ++ct) {
            const int n  = ct * 16 + lrow;
            const float b  = bias[n];
            float sc = 1.0f, sh = 0.0f;
            if constexpr (HAS_BN) { sc = scale[n] * scale_const; sh = shift[n]; }
            #pragma unroll
            for (int v = 0; v < 8; ++v) {
                float val = acc[ct][v] + b;
                if constexpr (HAS_BN) val = val * sc + sh;
                if constexpr (RELU)   val = fmaxf(val, 0.0f);
                Out[(size_t)(row_base + khalf * 8 + v) * HDIM + n] = val;
            }
        }
    } else {
        #pragma unroll
        for (int ct = 0; ct < 8; ++ct) {
            const int n  = ct * 16 + lrow;
            const float b  = bias[n];
            float sc = 1.0f, sh = 0.0f;
            if constexpr (HAS_BN) { sc = scale[n] * scale_const; sh = shift[n]; }
            #pragma unroll
            for (int v = 0; v < 8; ++v) {
                const int m = row_base + khalf * 8 + v;
                if (m < nrows) {
                    float val = acc[ct][v] + b;
                    if constexpr (HAS_BN) val = val * sc + sh;
                    if constexpr (RELU)   val = fmaxf(val, 0.0f);
                    Out[(size_t)m * HDIM + n] = val;
                }
            }
        }
    }
}

// ---------------------------------------------------------------------------
// segment_sum over edges: wave per edge; lane i handles feature chunk [4i..4i+3]
// agg[dst] += h[src]   (agg pre-zeroed)
// ---------------------------------------------------------------------------
__global__ __launch_bounds__(256) void scatter_add_edges(
    const float* __restrict__ h,
    const int* __restrict__ src, const int* __restrict__ dst,
    float* __restrict__ agg, int nedges)
{
    const int gid  = blockIdx.x * 256 + threadIdx.x;
    const int e    = gid >> 5;
    const int lane = gid & 31;
    if (e >= nedges) return;
    const int s = src[e];
    const int d = dst[e];
    const float4 val = ((const float4*)(h + (size_t)s * HDIM))[lane];
    float* o = agg + (size_t)d * HDIM + lane * 4;
    atomicAdd(o + 0, val.x);
    atomicAdd(o + 1, val.y);
    atomicAdd(o + 2, val.z);
    atomicAdd(o + 3, val.w);
}

// ---------------------------------------------------------------------------
// graph pooling: wave per node; out[batch[n]] += h[n]   (out pre-zeroed)
// ---------------------------------------------------------------------------
__global__ __launch_bounds__(256) void pool_nodes(
    const float* __restrict__ h, const int* __restrict__ batch,
    float* __restrict__ out, int nnodes)
{
    const int gid  = blockIdx.x * 256 + threadIdx.x;
    const int node = gid >> 5;
    const int lane = gid & 31;
    if (node >= nnodes) return;
    const int g = batch[node];
    const float4 val = ((const float4*)(h + (size_t)node * HDIM))[lane];
    float* o = out + (size_t)g * HDIM + lane * 4;
    atomicAdd(o + 0, val.x);
    atomicAdd(o + 1, val.y);
    atomicAdd(o + 2, val.z);
    atomicAdd(o + 3, val.w);
}

extern "C" void kernel_launch(void* const* d_in, const int* in_sizes, int n_in,
                              void* d_out, int out_size, void* d_ws, size_t ws_size,
                              hipStream_t stream) {
    const float* x     = (const float*)d_in[0];
    const int*   eidx  = (const int*)  d_in[1];
    const int*   batch = (const int*)  d_in[2];
    const float* t1_w  = (const float*)d_in[3];
    const float* t1_b  = (const float*)d_in[4];
    const float* t2_w  = (const float*)d_in[5];
    const float* t2_b  = (const float*)d_in[6];
    const float* W1    = (const float*)d_in[7];
    const float* b1    = (const float*)d_in[8];
    const float* gamma = (const float*)d_in[9];
    const float* beta  = (const float*)d_in[10];
    const float* W2    = (const float*)d_in[11];
    const float* b2    = (const float*)d_in[12];

    const int N = in_sizes[0] / HDIM;          // 100000
    const int E = in_sizes[1] / 2;             // 1600000
    const int L = in_sizes[7] / (HDIM * HDIM); // 3

    const int* src = eidx;
    const int* dst = eidx + E;

    float* h   = (float*)d_ws;                 // [N,128]
    float* agg = h   + (size_t)N * HDIM;       // [N,128]
    float* z   = agg + (size_t)N * HDIM;       // [N,128]

    const float bn_c = rsqrtf(1.0f + 1e-5f);   // BN eval-mode scale constant

    const dim3 blk(256);
    const dim3 gemm_grid((N + 127) / 128);
    const dim3 edge_grid((unsigned)(((size_t)E * 32 + 255) / 256));
    const dim3 node_grid((unsigned)(((size_t)N * 32 + 255) / 256));

    // h = relu(x @ t1_w^T + t1_b)
    gemm128_wmma<false, false, true><<<gemm_grid, blk, 0, stream>>>(
        x, nullptr, t1_w, t1_b, nullptr, nullptr, h, N, 1.0f);

    for (int l = 0; l < L; ++l) {
        hipMemsetAsync(agg, 0, (size_t)N * HDIM * sizeof(float), stream);
        scatter_add_edges<<<edge_grid, blk, 0, stream>>>(h, src, dst, agg, E);
        // z = relu(((h+agg) @ W1^T + b1) * bn_scale + beta)
        gemm128_wmma<true, true, true><<<gemm_grid, blk, 0, stream>>>(
            h, agg, W1 + (size_t)l * HDIM * HDIM, b1 + l * HDIM,
            gamma + l * HDIM, beta + l * HDIM, z, N, bn_c);
        // h = z @ W2^T + b2
        gemm128_wmma<false, false, false><<<gemm_grid, blk, 0, stream>>>(
            z, nullptr, W2 + (size_t)l * HDIM * HDIM, b2 + l * HDIM,
            nullptr, nullptr, h, N, 1.0f);
    }

    // h2 = h @ t2_w^T + t2_b   (reuse z)
    gemm128_wmma<false, false, false><<<gemm_grid, blk, 0, stream>>>(
        h, nullptr, t2_w, t2_b, nullptr, nullptr, z, N, 1.0f);

    // graph_emb = segment_sum(h2, batch, G)
    hipMemsetAsync(d_out, 0, (size_t)out_size * sizeof(float), stream);
    pool_nodes<<<node_grid, blk, 0, stream>>>(z, batch, (float*)d_out, N);
}